// CharacterLanguageModel_42219528520022
// MI455X (gfx1250) — compile-verified
//
#include <hip/hip_runtime.h>
#include <hip/hip_bf16.h>
#include <stdint.h>
#include <stddef.h>

// ---------------------------------------------------------------------------
// CharacterLanguageModel: ragged gather + segment-mean.
//
// Roofline: ~0.5 GFLOP vs ~0.39 GB min traffic -> pure HBM-bandwidth problem
// (~17us floor at 23.3 TB/s). WMMA via one-hot would cost ~205 GFLOP of matrix
// work for the same memory traffic -> strictly worse. The CDNA5 path that pays
// off here is async global->LDS staging of the 51.2 KB embedding table
// (GLOBAL_LOAD_ASYNC_TO_LDS_B128 + S_WAIT_ASYNCCNT), so the ~1.6 GB of random
// 200 B row gathers are served from LDS at register-like bandwidth.
//
// Single pass, no atomics: segment_ids is sorted, so each word is a contiguous
// run. A thread owns the words that *start* inside its 32-char chunk, scans
// each run to its end (possibly past the chunk), accumulates 25x float2 in
// VGPRs (v_pk_add_f32), scales by 1/count and stores. Gap (empty) words are
// zeroed by the owner of the next present word; the tail past the last present
// word is zeroed by the owner of the final run. Every output row is written
// exactly once per call (harness poisons d_out and never re-poisons).
// ---------------------------------------------------------------------------

typedef __attribute__((ext_vector_type(2))) float f32x2;

#define EMB        50          // embedding size (floats)
#define EMBV       25          // float2 per row
#define TBL_ROWS   256
#define TBL_DWORDS (TBL_ROWS * EMB)   // 12800 dwords = 51200 B
#define TBL_B128   (TBL_DWORDS / 4)   // 3200 16-byte transfers
#define BLOCK      256
#define CHUNK      32          // chars per thread

__global__ __launch_bounds__(BLOCK)
void cls_segmean_kernel(const int* __restrict__ char_ids,
                        const int* __restrict__ seg_ids,
                        const float* __restrict__ table,
                        float* __restrict__ out,
                        int n_chars, int num_words)
{
    __shared__ float s_tab[TBL_DWORDS];   // 51.2 KB of 320 KB/WGP

    // ---- Stage embedding table into LDS via CDNA5 async copies (ASYNCcnt) --
    {
        const int tid = threadIdx.x;
        // Low 32 bits of the generic address of an LDS object are the LDS
        // byte offset (ISA 10.2); the async instruction
        // takes that offset in a single VGPR (VDST).
        const unsigned lds_base = (unsigned)(size_t)(void*)s_tab;
#pragma unroll
        for (int k = 0; k < (TBL_B128 + BLOCK - 1) / BLOCK; ++k) {
            int e = k * BLOCK + tid;
            if (e < TBL_B128) {
                unsigned lds_addr = lds_base + (unsigned)e * 16u;
                unsigned goff     = (unsigned)e * 16u;   // GVS: SGPR base + VGPR offset
                asm volatile(
                    "global_load_async_to_lds_b128 %0, %1, %2"
                    :
                    : "v"(lds_addr), "v"(goff), "s"(table)
                    : "memory");
            }
        }
#if defined(__has_builtin) && __has_builtin(__builtin_amdgcn_s_wait_asynccnt)
        __builtin_amdgcn_s_wait_asynccnt(0);
#else
        asm volatile("s_wait_asynccnt 0x0" ::: "memory");
#endif
        __syncthreads();   // all 8 waves' slices visible block-wide
    }

    // ---- Ownership scan over this thread's contiguous char chunk -----------
    const int gthread = blockIdx.x * BLOCK + threadIdx.x;
    int t0 = gthread * CHUNK;
    if (t0 >= n_chars) return;
    int t1 = t0 + CHUNK;
    if (t1 > n_chars) t1 = n_chars;

    int prev = (t0 > 0) ? __builtin_nontemporal_load(seg_ids + (t0 - 1)) : -1;
    int i = t0;
    while (i < t1) {
        int s = __builtin_nontemporal_load(seg_ids + i);
        if (s == prev) { ++i; continue; }   // continuation of a word owned upstream

        // We own word s. First zero any empty words in the gap (prev, s).
        for (int w = prev + 1; w < s; ++w) {
            f32x2* zp = (f32x2*)(out + (size_t)w * EMB);
#pragma unroll
            for (int k = 0; k < EMBV; ++k) {
                f32x2 z = {0.0f, 0.0f};
                __builtin_nontemporal_store(z, zp + k);
            }
        }

        // Accumulate the full run [i, j) of word s (may extend past t1).
        f32x2 acc[EMBV] = {};
        int cnt = 0;
        int j = i;
        while (j < n_chars && __builtin_nontemporal_load(seg_ids + j) == s) {
            int c = __builtin_nontemporal_load(char_ids + j);
            const f32x2* row = (const f32x2*)&s_tab[c * EMB];  // 8B-aligned LDS row
#pragma unroll
            for (int k = 0; k < EMBV; ++k) acc[k] += row[k];   // v_pk_add_f32
            ++cnt; ++j;
        }

        const float inv = 1.0f / (float)cnt;   // cnt >= 1 by construction
        f32x2* op = (f32x2*)(out + (size_t)s * EMB);
#pragma unroll
        for (int k = 0; k < EMBV; ++k) {
            f32x2 r = acc[k] * inv;
            __builtin_nontemporal_store(r, op + k);
        }

        if (j >= n_chars) {
            // Owner of the final run zeroes everything past the last word id.
            for (int w = s + 1; w < num_words; ++w) {
                f32x2* zp = (f32x2*)(out + (size_t)w * EMB);
#pragma unroll
                for (int k = 0; k < EMBV; ++k) {
                    f32x2 z = {0.0f, 0.0f};
                    __builtin_nontemporal_store(z, zp + k);
                }
            }
        }
        prev = s;
        i = j;
    }
}

extern "C" void kernel_launch(void* const* d_in, const int* in_sizes, int n_in,
                              void* d_out, int out_size, void* d_ws, size_t ws_size,
                              hipStream_t stream)
{
    (void)n_in; (void)d_ws; (void)ws_size;
    const int*   char_ids = (const int*)d_in[0];
    const int*   seg_ids  = (const int*)d_in[1];
    // d_in[2] = num_words scalar (unused; derived from out_size)
    const float* table    = (const float*)d_in[3];
    float*       out      = (float*)d_out;

    const int n_chars   = in_sizes[0];
    const int num_words = out_size / EMB;

    const int n_threads = (n_chars + CHUNK - 1) / CHUNK;     // 250,000
    const int blocks    = (n_threads + BLOCK - 1) / BLOCK;   // ~977

    cls_segmean_kernel<<<blocks, BLOCK, 0, stream>>>(
        char_ids, seg_ids, table, out, n_chars, num_words);
}